// GraphAttentionV2Layer_79551384256743
// MI455X (gfx1250) — compile-verified
//
#include <hip/hip_runtime.h>
#include <hip/hip_bf16.h>
#include <math.h>

typedef __attribute__((ext_vector_type(2))) float v2f;
typedef __attribute__((ext_vector_type(8))) float v8f;
typedef unsigned int v4u __attribute__((ext_vector_type(4)));
typedef int          v8i __attribute__((ext_vector_type(8)));
typedef int          v4i __attribute__((ext_vector_type(4)));

#define NB    4
#define NN    512
#define FIN   256
#define NH    8
#define NF    32
#define HF    256          // NH*NF
#define MROWS 2048         // NB*NN
#define NEG_SLOPE 0.21f
#define MASK_VAL  (-1000000.0f)

// ---------------------------------------------------------------------------
// Kernel 1: g_l = h @ W_l (stored transposed per (b,h): glT[((b*8+h)*32+f)*512+j])
//           g_r = h @ W_r (row-major: gr[(b*512+j)*256 + h*32+f])
// One wave per 16x16 output tile, fp32 WMMA 16x16x4, K=256 -> 64 steps,
// both GEMMs share the A loads (128 WMMAs per wave).
// ---------------------------------------------------------------------------
__global__ __launch_bounds__(256) void gat_gemm_glr(
    const float* __restrict__ h,
    const float* __restrict__ Wl,
    const float* __restrict__ Wr,
    float* __restrict__ glT,
    float* __restrict__ gr)
{
    const int lane = threadIdx.x & 31;
    const int wave = threadIdx.x >> 5;
    const int tile = blockIdx.x * 8 + wave;     // 2048 tiles total
    const int rowTile = tile >> 4;              // 0..127  (rows of 16)
    const int colTile = tile & 15;              // 0..15   (cols of 16)

    const int ln  = lane & 15;
    const int hi  = lane >> 4;                  // 0 or 1
    const int m   = rowTile * 16 + ln;          // A row for this lane
    const int n   = colTile * 16 + ln;          // B/D col for this lane
    const int kof = hi * 2;                     // K sub-offset (0 or 2)

    v8f accL = {};
    v8f accR = {};

    const float* hrow = h + (size_t)m * FIN;
    for (int k = 0; k < FIN; k += 4) {
        const int kr = k + kof;
        v2f a = *(const v2f*)(hrow + kr);       // A: lane holds h[m][kr], h[m][kr+1]
        v2f bl, br;
        bl.x = Wl[(size_t)(kr + 0) * HF + n];
        bl.y = Wl[(size_t)(kr + 1) * HF + n];
        br.x = Wr[(size_t)(kr + 0) * HF + n];
        br.y = Wr[(size_t)(kr + 1) * HF + n];
        accL = __builtin_amdgcn_wmma_f32_16x16x4_f32(false, a, false, bl,
                                                     (short)0, accL, false, false);
        accR = __builtin_amdgcn_wmma_f32_16x16x4_f32(false, a, false, br,
                                                     (short)0, accR, false, false);
    }

    // D layout: VGPR v holds row (v + 8*hi), col n
    const int hh = n >> 5;                      // head
    const int ff = n & 31;                      // feature within head
    #pragma unroll
    for (int v = 0; v < 8; ++v) {
        const int row = rowTile * 16 + hi * 8 + v;   // global row 0..2047
        const int bb  = row >> 9;
        const int j   = row & 511;
        glT[((size_t)((bb * NH + hh) * NF + ff)) * NN + j] = accL[v];
        gr [(size_t)row * HF + n]                          = accR[v];
    }
}

// ---------------------------------------------------------------------------
// Kernel 2: fused attention. One block = 4 waves = one (b,h,64-row i-band);
// each wave owns a 16-row i-tile.
//   - TDM (tensor_load_to_lds) stages the 64KB glT (b,h) slice into LDS once.
//   - e[i,j] = sum_f w[f]*leaky(glT[f][j] + gr[i][f]) on VALU (ds reads,
//     conflict-free: lane == consecutive j), mask by adj, shfl softmax.
//   - out tile = (1/sum) * exp_row @ gr via fp32 WMMA 16x16x4, K=512.
// LDS: 64KB glT_s + 129KB a_s + 256B = ~193KB (one workgroup per WGP).
// ---------------------------------------------------------------------------
__global__ __launch_bounds__(128) void gat_attn(
    const float* __restrict__ glT,
    const float* __restrict__ gr,
    const int*   __restrict__ adj,
    const float* __restrict__ attn_w,
    float* __restrict__ out)
{
    __shared__ float glT_s[NF][NN];   // 64 KB, [f][j]
    __shared__ float a_s[64][516];    // wave w owns rows [16w, 16w+16)
    __shared__ float r_s[64];         // per-row 1/sum

    const int bid  = blockIdx.x;      // 4*8*8 = 256 blocks
    const int ibB  = bid & 7;         // 64-row band
    const int hh   = (bid >> 3) & 7;  // head
    const int bb   = bid >> 6;        // batch

    const int tid  = threadIdx.x;
    const int wave = tid >> 5;
    const int lane = tid & 31;
    const int ln   = lane & 15;
    const int hi   = lane >> 4;
    const int itile = ibB * 4 + wave; // this wave's 16-row tile within N

    const float* glTbase = glT + (size_t)((bb * NH + hh) * NF) * NN; // [f][j], 64KB contig
    const float* grbase  = gr  + (size_t)(bb * NN) * HF + hh * NF;   // row j: grbase + j*HF

    // ---- TDM stage-in of glT slice (wave 0 issues one descriptor) ----
    if (tid < 32) {
        const unsigned ldsAddr = (unsigned)(uintptr_t)(&glT_s[0][0]); // LDS offset = low 32 bits
        const unsigned long long ga = (unsigned long long)(uintptr_t)glTbase;
        v4u g0;
        g0.x = 1u;                                        // count=1, user descriptor
        g0.y = ldsAddr;                                   // lds_addr
        g0.z = (unsigned)(ga & 0xFFFFFFFFull);            // global_addr[31:0]
        g0.w = (unsigned)((ga >> 32) & 0x1FFFFFFull)      // global_addr[56:32]
             | (2u << 30);                                // type=2 ("image")
        v8i g1;
        g1[0] = 0x00020000;                               // data_size=2 -> 4 bytes
        g1[1] = (int)(16384u << 16);                      // tensor_dim0 = 16384 (lo16)
        g1[2] = (int)(1u << 16);                          // dim0 hi=0, tensor_dim1 = 1
        g1[3] = (int)(16384u << 16);                      // dim1 hi=0, tile_dim0 = 16384
        g1[4] = 0;                                        // tile_dim1=0, tile_dim2=0
        g1[5] = 16384;                                    // tensor_dim0_stride
        g1[6] = 0;
        g1[7] = 0;
        v4i gz4 = {0, 0, 0, 0};                           // groups 2/3 unused (<=2D)
        v8i gz8 = {0, 0, 0, 0, 0, 0, 0, 0};               // extra group (6-arg toolchain)
        __builtin_amdgcn_tensor_load_to_lds(g0, g1, gz4, gz4, gz8, 0);
        __builtin_amdgcn_s_wait_tensorcnt(0);
    }
    __syncthreads();

    // attention weights -> uniform registers
    float wv[NF];
    #pragma unroll
    for (int f = 0; f < NF; ++f) wv[f] = attn_w[f];

    // ---- scores + softmax for this wave's 16 rows ----
    for (int r = 0; r < 16; ++r) {
        const int i  = itile * 16 + r;            // query row in [0,512)
        const int ar = wave * 16 + r;             // a_s row
        float gri[NF];
        const float* grow = grbase + (size_t)i * HF;   // uniform address
        #pragma unroll
        for (int f = 0; f < NF; ++f) gri[f] = grow[f];

        float lmax = -3.0e38f;
        for (int jc = 0; jc < 16; ++jc) {
            const int j = jc * 32 + lane;
            float acc = 0.0f;
            #pragma unroll
            for (int f = 0; f < NF; ++f) {
                float x   = glT_s[f][j] + gri[f];      // ds_load_b32, no conflicts
                float act = (x >= 0.0f) ? x : NEG_SLOPE * x;
                acc = fmaf(wv[f], act, acc);
            }
            const int ad = adj[(size_t)i * NN + j];
            float e = (ad == 0) ? MASK_VAL : acc;
            a_s[ar][j] = e;
            lmax = fmaxf(lmax, e);
        }
        #pragma unroll
        for (int off = 16; off >= 1; off >>= 1)
            lmax = fmaxf(lmax, __shfl_xor(lmax, off, 32));

        float lsum = 0.0f;
        for (int jc = 0; jc < 16; ++jc) {
            const int j = jc * 32 + lane;
            float ex = __expf(a_s[ar][j] - lmax);
            a_s[ar][j] = ex;
            lsum += ex;
        }
        #pragma unroll
        for (int off = 16; off >= 1; off >>= 1)
            lsum += __shfl_xor(lsum, off, 32);
        if (lane == 0) r_s[wave * 16 + r] = 1.0f / lsum;
    }
    // No barrier needed: WMMA below reads only this wave's a_s/r_s rows.

    // ---- attn_res tile: (16 x 512) @ (512 x 32) via WMMA, two 16-col tiles ----
    const int kof = hi * 2;
    #pragma unroll
    for (int ft = 0; ft < 2; ++ft) {
        const int nf = ft * 16 + ln;              // output feature 0..31
        v8f c = {};
        for (int k = 0; k < NN; k += 4) {
            const int kr = k + kof;
            v2f a = *(const v2f*)&a_s[wave * 16 + ln][kr];  // ds_load_b64, 8B aligned
            v2f b;
            b.x = grbase[(size_t)(kr + 0) * HF + nf];
            b.y = grbase[(size_t)(kr + 1) * HF + nf];
            c = __builtin_amdgcn_wmma_f32_16x16x4_f32(false, a, false, b,
                                                      (short)0, c, false, false);
        }
        #pragma unroll
        for (int v = 0; v < 8; ++v) {
            const int row = hi * 8 + v;           // row within i-tile
            const int i   = itile * 16 + row;
            out[((size_t)(bb * NN + i)) * HF + hh * NF + nf] = c[v] * r_s[wave * 16 + row];
        }
    }
}

// ---------------------------------------------------------------------------
extern "C" void kernel_launch(void* const* d_in, const int* in_sizes, int n_in,
                              void* d_out, int out_size, void* d_ws, size_t ws_size,
                              hipStream_t stream) {
    const float* h   = (const float*)d_in[0];
    const int*   adj = (const int*)  d_in[1];
    const float* Wl  = (const float*)d_in[2];
    const float* Wr  = (const float*)d_in[3];
    const float* aw  = (const float*)d_in[4];
    float*       outp = (float*)d_out;

    float* glT = (float*)d_ws;                      // 2048*256 floats = 2 MB
    float* gr  = glT + (size_t)MROWS * HF;          // 2 MB more

    gat_gemm_glr<<<256, 256, 0, stream>>>(h, Wl, Wr, glT, gr);
    gat_attn<<<NB * NH * (NN / 64), 128, 0, stream>>>(glT, gr, adj, aw, outp);
}